// GCNEncoder_26577257628042
// MI455X (gfx1250) — compile-verified
//
#include <hip/hip_runtime.h>
#include <hip/hip_bf16.h>

// ---------------------------------------------------------------------------
// GCN 2-layer forward for MI455X (gfx1250), fp32.
//   h1  = X @ W1 + b1                  (WMMA f32 16x16x4)
//   a1  = D^-1/2 A_hat D^-1/2 h1       (self-loop init + edge atomic scatter)
//   h2  = relu(a1) @ W2 + b2           (WMMA, relu fused into A load)
//   out = D^-1/2 A_hat D^-1/2 h2
// All buffers (~82 MB) are L2-resident (192 MB HBM-backed L2), so the edge
// scatter is bound by L2 fp32-atomic throughput; we force the hardware
// global_atomic_add_f32 (device scope) instead of risking a CAS expansion.
// ---------------------------------------------------------------------------

typedef __attribute__((ext_vector_type(2))) float v2f;
typedef __attribute__((ext_vector_type(8))) float v8f;

// Hardware fp32 atomic add, non-returning, device scope (coherent across
// WGPs via L2 atomic units). Tracked with STOREcnt; s_endpgm's implicit
// wait-idle covers completion at kernel end.
__device__ __forceinline__ void atomic_add_f32_dev(float* p, float v) {
    asm volatile("global_atomic_add_f32 %0, %1, off scope:SCOPE_DEV"
                 :: "v"(p), "v"(v)
                 : "memory");
}

// ---------------- small utility kernels ----------------

__global__ void fill_f32(float* __restrict__ p, float v, int n) {
    int i = blockIdx.x * blockDim.x + threadIdx.x;
    if (i < n) p[i] = v;
}

__global__ void deg_count(const long long* __restrict__ col, float* __restrict__ deg, int E) {
    int e = blockIdx.x * blockDim.x + threadIdx.x;
    if (e < E) {
        int c = (int)col[e];
        atomic_add_f32_dev(&deg[c], 1.0f);
    }
}

__global__ void rsqrt_inplace(float* __restrict__ p, int n) {
    int i = blockIdx.x * blockDim.x + threadIdx.x;
    if (i < n) p[i] = rsqrtf(p[i]);   // deg >= 1 always (self loops)
}

// out[i,j] = dinv[i]^2 * h[i,j]  -- self-loop term; also initializes buffer.
__global__ void selfloop_scale(const float* __restrict__ dinv, const float* __restrict__ h,
                               float* __restrict__ out, long long total, int shiftC) {
    long long idx = (long long)blockIdx.x * blockDim.x + threadIdx.x;
    if (idx >= total) return;
    int i = (int)(idx >> shiftC);
    float w = dinv[i];
    out[idx] = w * w * h[idx];
}

// ---------------- edge gather-scale-scatter ----------------
// One wave per edge (grid-stride). Row gather is fully coalesced (32 lanes x
// 16B float4 = contiguous C*4 bytes). Scatter uses hardware fp32 atomics into
// the L2-resident destination. The next iteration's full source row is
// prefetched (global_prefetch_b8).
__global__ void edge_aggregate(const long long* __restrict__ row,
                               const long long* __restrict__ col,
                               const float* __restrict__ dinv,
                               const float* __restrict__ h,
                               float* __restrict__ out,
                               int E, int C) {
    const int lane = threadIdx.x & 31;
    const int wavesPerBlock = blockDim.x >> 5;
    const int nwaves = gridDim.x * wavesPerBlock;
    const int floatsPerLane = C >> 5;     // 8 (C=256) or 4 (C=128)
    int wid = blockIdx.x * wavesPerBlock + (threadIdx.x >> 5);

    for (int e = wid; e < E; e += nwaves) {
        int en = e + nwaves;
        if (en < E) {
            // each lane prefetches its own slot -> covers the whole next row
            __builtin_prefetch(h + (size_t)row[en] * C + lane * floatsPerLane, 0, 1);
        }
        int r = (int)row[e];
        int c = (int)col[e];
        float w = dinv[r] * dinv[c];
        const float4* __restrict__ src = (const float4*)(h + (size_t)r * C);
        float* __restrict__ dst = out + (size_t)c * C;
        for (int v = lane; v * 4 < C; v += 32) {
            float4 d = src[v];
            float* p = dst + 4 * v;
            atomic_add_f32_dev(p + 0, w * d.x);
            atomic_add_f32_dev(p + 1, w * d.y);
            atomic_add_f32_dev(p + 2, w * d.z);
            atomic_add_f32_dev(p + 3, w * d.w);
        }
    }
}

// ---------------- WMMA fp32 GEMM:  D = opA(A[M,K]) @ B[K,N] + bias ----------
// One wave -> one 16x16 output tile, K swept 4 at a time with
// V_WMMA_F32_16X16X4_F32 (unroll 8 -> clause-batched loads feeding a WMMA
// burst, per round-1 codegen).
//
// Fragment layout (ISA 7.12.2, 32-bit):
//   A 16x4 : lanes 0-15 hold rows M=0..15 at K={k,k+1} (v0,v1);
//            lanes 16-31 same rows at K={k+2,k+3}.
//   B 4x16 : v0 = B[k][n] (lanes 0-15) / B[k+2][n] (lanes 16-31); v1 = k+1/k+3.
//   C/D    : vi, lane L -> D[m0 + i + 8*(L>>4)][n0 + (L&15)].
template <bool RELU>
__global__ void gemm_wmma_f32(const float* __restrict__ A, const float* __restrict__ B,
                              const float* __restrict__ bias, float* __restrict__ D,
                              int M, int K, int N) {
    const int lane = threadIdx.x & 31;
    const int wave = threadIdx.x >> 5;
    const int tilesN = N >> 4;
    const int tilesM = M >> 4;
    int tile = blockIdx.x * (blockDim.x >> 5) + wave;
    if (tile >= tilesM * tilesN) return;
    const int mt = tile / tilesN;
    const int nt = tile - mt * tilesN;
    const int m0 = mt << 4;
    const int n0 = nt << 4;
    const int half = lane >> 4;     // 0 for lanes 0-15, 1 for lanes 16-31
    const int l    = lane & 15;

    // seed accumulator with bias (bias depends only on column = l)
    const float bv = bias[n0 + l];
    v8f c;
#pragma unroll
    for (int i = 0; i < 8; ++i) c[i] = bv;

    const float* __restrict__ Arow = A + (size_t)(m0 + l) * K + 2 * half;
    const float* __restrict__ Bcol = B + (size_t)(2 * half) * N + n0 + l;

#pragma unroll 8
    for (int k = 0; k < K; k += 4) {
        v2f a, b;
        float a0 = Arow[k];
        float a1 = Arow[k + 1];
        if (RELU) {
            a0 = fmaxf(a0, 0.0f);
            a1 = fmaxf(a1, 0.0f);
        }
        a[0] = a0;
        a[1] = a1;
        b[0] = Bcol[(size_t)k * N];
        b[1] = Bcol[(size_t)(k + 1) * N];
        c = __builtin_amdgcn_wmma_f32_16x16x4_f32(
            /*neg_a=*/false, a, /*neg_b=*/false, b,
            /*c_mod=*/(short)0, c, /*reuse_a=*/false, /*reuse_b=*/false);
    }

#pragma unroll
    for (int i = 0; i < 8; ++i)
        D[(size_t)(m0 + i + 8 * half) * N + n0 + l] = c[i];
}

// ---------------------------------------------------------------------------

static inline int cdiv(long long a, long long b) { return (int)((a + b - 1) / b); }

extern "C" void kernel_launch(void* const* d_in, const int* in_sizes, int n_in,
                              void* d_out, int out_size, void* d_ws, size_t ws_size,
                              hipStream_t stream) {
    const float*     x   = (const float*)d_in[0];
    const long long* ei  = (const long long*)d_in[1];   // int64 edge_index [2, E]
    const float*     W1  = (const float*)d_in[2];
    const float*     b1  = (const float*)d_in[3];
    const float*     W2  = (const float*)d_in[4];
    const float*     b2  = (const float*)d_in[5];
    float*           out = (float*)d_out;

    const int in_c  = 256;
    const int hid   = in_sizes[3];        // 256
    const int out_c = in_sizes[5];        // 128
    const int N     = in_sizes[0] / in_c; // 20000
    const int E     = in_sizes[1] / 2;    // 640000

    const long long* row = ei;
    const long long* col = ei + E;

    // workspace layout (256B aligned)
    char*  ws  = (char*)d_ws;
    size_t off = 0;
    auto carve = [&](size_t bytes) -> void* {
        void* p = ws + off;
        off = (off + bytes + 255) & ~(size_t)255;
        return p;
    };
    float* dinv = (float*)carve((size_t)N * 4);
    float* h1   = (float*)carve((size_t)N * hid * 4);
    float* agg1 = (float*)carve((size_t)N * hid * 4);
    float* h2   = (float*)carve((size_t)N * out_c * 4);
    (void)ws_size;

    const int BLK = 256;                  // 8 waves per block

    // --- degree norm: deg = 1 (self loop) + count(col); dinv = deg^-1/2
    fill_f32<<<cdiv(N, BLK), BLK, 0, stream>>>(dinv, 1.0f, N);
    deg_count<<<cdiv(E, BLK), BLK, 0, stream>>>(col, dinv, E);
    rsqrt_inplace<<<cdiv(N, BLK), BLK, 0, stream>>>(dinv, N);

    // --- layer 1 linear: h1 = X @ W1 + b1
    {
        int tiles = (N / 16) * (hid / 16);
        gemm_wmma_f32<false><<<cdiv(tiles, 8), BLK, 0, stream>>>(x, W1, b1, h1, N, in_c, hid);
    }
    // --- layer 1 aggregate: agg1 = dinv^2*h1 (init) + edge scatter
    {
        long long total = (long long)N * hid;
        selfloop_scale<<<cdiv(total, BLK), BLK, 0, stream>>>(dinv, h1, agg1, total, 8 /*log2(hid)*/);
        edge_aggregate<<<20000, BLK, 0, stream>>>(row, col, dinv, h1, agg1, E, hid);
    }
    // --- layer 2 linear (relu fused): h2 = relu(agg1) @ W2 + b2
    {
        int tiles = (N / 16) * (out_c / 16);
        gemm_wmma_f32<true><<<cdiv(tiles, 8), BLK, 0, stream>>>(agg1, W2, b2, h2, N, hid, out_c);
    }
    // --- layer 2 aggregate into d_out
    {
        long long total = (long long)N * out_c;
        selfloop_scale<<<cdiv(total, BLK), BLK, 0, stream>>>(dinv, h2, out, total, 7 /*log2(out_c)*/);
        edge_aggregate<<<20000, BLK, 0, stream>>>(row, col, dinv, h2, out, E, out_c);
    }
}